// LinearSpatial_30133490549252
// MI455X (gfx1250) — compile-verified
//
#include <hip/hip_runtime.h>
#include <hip/hip_bf16.h>
#include <stdint.h>

typedef __attribute__((ext_vector_type(2))) float v2f;
typedef __attribute__((ext_vector_type(8))) float v8f;

#define F_IN  256
#define F_OUT 128

// ---------------------------------------------------------------------------
// 1) deg[i] = 1.0f  (self loop contribution)
// ---------------------------------------------------------------------------
__global__ void init_deg(float* __restrict__ deg, int n) {
    int i = blockIdx.x * blockDim.x + threadIdx.x;
    if (i < n) deg[i] = 1.0f;
}

// ---------------------------------------------------------------------------
// 2) deg[dst[e]] += 1  (f32 atomic: exact for counts < 2^24)
//    edge_index is streamed once -> non-temporal load keeps it out of L2.
// ---------------------------------------------------------------------------
__global__ void count_deg(const long long* __restrict__ dst,
                          float* __restrict__ deg, int E) {
    int e = blockIdx.x * blockDim.x + threadIdx.x;
    if (e < E) {
        int d = (int)__builtin_nontemporal_load(dst + e);
        atomicAdd(&deg[d], 1.0f);
    }
}

// ---------------------------------------------------------------------------
// 3) deg -> rsqrt(deg), in place
// ---------------------------------------------------------------------------
__global__ void rsqrt_deg(float* __restrict__ deg, int n) {
    int i = blockIdx.x * blockDim.x + threadIdx.x;
    if (i < n) deg[i] = rsqrtf(deg[i]);
}

// ---------------------------------------------------------------------------
// 4) h = x @ W   via V_WMMA_F32_16X16X4_F32 (full fp32, matches reference)
//    Block = 256 threads = 8 waves. Block b owns output rows [16b,16b+16);
//    wave w owns output cols [16w, 16w+16). K-loop: 256/4 = 64 WMMAs.
//
//    A (16x4 f32, 2 VGPRs): lanes 0-15 hold rows m0+l, v0=K+0, v1=K+1;
//                           lanes16-31 hold rows m0+l, v0=K+2, v1=K+3.
//    B (4x16 f32, 2 VGPRs): symmetric layout, N striped over lane%16.
//    C/D (16x16 f32, 8 VGPRs): vgpr r, lane l -> row r+8*(l/16), col l%16.
//
//    x is 102 MB read-once -> NT loads so it does not evict h/out from L2.
//    h stores stay RT: the scatter kernel re-reads h from L2.
// ---------------------------------------------------------------------------
__global__ void __launch_bounds__(256) gemm_xw(const float* __restrict__ x,
                                               const float* __restrict__ W,
                                               float* __restrict__ h) {
    const int wave  = threadIdx.x >> 5;   // 0..7 -> n-tile
    const int lane  = threadIdx.x & 31;
    const int lhalf = lane >> 4;          // 0 or 1
    const int l15   = lane & 15;
    const int m0 = blockIdx.x * 16;
    const int n0 = wave * 16;

    // per-lane base pointers (K offset of +2 for the upper lane half)
    const float* __restrict__ arow = x + (size_t)(m0 + l15) * F_IN + 2 * lhalf;
    const float* __restrict__ bcol = W + (size_t)(2 * lhalf) * F_OUT + n0 + l15;

    v8f c = {};
    #pragma unroll 8
    for (int k = 0; k < F_IN; k += 4) {
        v2f a = __builtin_nontemporal_load((const v2f*)(arow + k)); // K+0,K+1 (or +2,+3)
        v2f b = { bcol[(size_t)k * F_OUT],
                  bcol[(size_t)(k + 1) * F_OUT] };                  // rows K.. of W
        c = __builtin_amdgcn_wmma_f32_16x16x4_f32(
                /*neg_a=*/false, a, /*neg_b=*/false, b,
                /*c_mod=*/(short)0, c, /*reuse_a=*/false, /*reuse_b=*/false);
    }

    float* __restrict__ hout = h + (size_t)(m0 + 8 * lhalf) * F_OUT + n0 + l15;
    #pragma unroll
    for (int r = 0; r < 8; ++r)
        hout[(size_t)r * F_OUT] = c[r];
}

// ---------------------------------------------------------------------------
// 5) out = b + h * dinv^2   (self-loop term + bias; initializes poisoned d_out)
//    One thread per (node, col/4); float4 vectorized. out stays RT (L2) for
//    the atomic scatter that follows.
// ---------------------------------------------------------------------------
__global__ void __launch_bounds__(256) self_bias(const float* __restrict__ h,
                                                 const float* __restrict__ dinv,
                                                 const float* __restrict__ b,
                                                 float* __restrict__ out, int n) {
    int idx  = blockIdx.x * blockDim.x + threadIdx.x;
    int node = idx >> 5;            // 32 float4 groups per node (128 cols)
    int c4   = (idx & 31) * 4;
    if (node >= n) return;
    float w = dinv[node];
    w = w * w;
    float4 hv = *(const float4*)(h + (size_t)node * F_OUT + c4);
    float4 bv = *(const float4*)(b + c4);
    float4 o;
    o.x = fmaf(hv.x, w, bv.x);
    o.y = fmaf(hv.y, w, bv.y);
    o.z = fmaf(hv.z, w, bv.z);
    o.w = fmaf(hv.w, w, bv.w);
    *(float4*)(out + (size_t)node * F_OUT + c4) = o;
}

// ---------------------------------------------------------------------------
// 6) edge scatter: out[dst] += h[src] * dinv[src]*dinv[dst]
//    One wave32 per edge: lane l handles cols [4l, 4l+4).
//    - edge_index loads are NT (read-once stream, keep out of L2)
//    - s/d are wave-uniform -> readfirstlane puts them in SGPRs so the
//      h gather / out atomics use scalar base + lane offset
//    - h/out are L2-resident (51.2 MB each vs 192 MB L2): gathers and the
//      global_atomic_add_f32 RMWs stay on-chip.
// ---------------------------------------------------------------------------
__global__ void __launch_bounds__(256) scatter_edges(const long long* __restrict__ src,
                                                     const long long* __restrict__ dst,
                                                     const float* __restrict__ h,
                                                     const float* __restrict__ dinv,
                                                     float* __restrict__ out, int E) {
    int t    = blockIdx.x * blockDim.x + threadIdx.x;
    int e    = t >> 5;              // one wave32 per edge (uniform across wave)
    int lane = t & 31;
    if (e >= E) return;
    int s = (int)__builtin_nontemporal_load(src + e);
    int d = (int)__builtin_nontemporal_load(dst + e);
    s = __builtin_amdgcn_readfirstlane(s);   // wave-uniform -> SGPR
    d = __builtin_amdgcn_readfirstlane(d);
    float w = dinv[s] * dinv[d];
    float4 hv = *(const float4*)(h + (size_t)s * F_OUT + lane * 4);
    float* op = out + (size_t)d * F_OUT + lane * 4;
    atomicAdd(op + 0, hv.x * w);
    atomicAdd(op + 1, hv.y * w);
    atomicAdd(op + 2, hv.z * w);
    atomicAdd(op + 3, hv.w * w);
}

// ---------------------------------------------------------------------------
extern "C" void kernel_launch(void* const* d_in, const int* in_sizes, int n_in,
                              void* d_out, int out_size, void* d_ws, size_t ws_size,
                              hipStream_t stream) {
    const float*     x  = (const float*)d_in[0];
    const long long* ei = (const long long*)d_in[1];   // int64 edge_index [2, E]
    const float*     W  = (const float*)d_in[2];
    const float*     b  = (const float*)d_in[3];
    float*           out = (float*)d_out;

    const int N = in_sizes[0] / F_IN;       // 100000
    const int E = in_sizes[1] / 2;          // 1600000
    const long long* src = ei;
    const long long* dst = ei + E;

    // workspace layout: [ dinv: N floats | h: N*F_OUT floats ]
    char* ws = (char*)d_ws;
    float* dinv = (float*)ws;
    size_t off  = ((size_t)N * sizeof(float) + 255) & ~(size_t)255;
    float* h    = (float*)(ws + off);

    // 1) deg = 1
    init_deg<<<(N + 255) / 256, 256, 0, stream>>>(dinv, N);
    // 2) deg += per-edge counts
    count_deg<<<(E + 255) / 256, 256, 0, stream>>>(dst, dinv, E);
    // 3) deg -> rsqrt(deg)
    rsqrt_deg<<<(N + 255) / 256, 256, 0, stream>>>(dinv, N);
    // 4) h = x @ W  (WMMA fp32)
    gemm_xw<<<N / 16, 256, 0, stream>>>(x, W, h);
    // 5) out = b + h * dinv^2
    {
        long long threads = (long long)N * 32;
        self_bias<<<(unsigned)((threads + 255) / 256), 256, 0, stream>>>(h, dinv, b, out, N);
    }
    // 6) scatter edges
    {
        long long threads = (long long)E * 32;
        scatter_edges<<<(unsigned)((threads + 255) / 256), 256, 0, stream>>>(src, dst, h, dinv, out, E);
    }
}